// AttitudeController_88845693485856
// MI455X (gfx1250) — compile-verified
//
#include <hip/hip_runtime.h>

// Quadrotor attitude controller, B = 2^21 elements.
// HBM-bound (~84 B/elem, ~300 flop/elem) -> optimize data movement:
//  - per-block async DMA of the AoS slabs into LDS (gfx1250 GLOBAL_LOAD_ASYNC_TO_LDS_B128,
//    tracked by ASYNCcnt, drained with s_wait_asynccnt) — straight-line, no dynamic loops
//  - conflict-free stride-13 LDS reads, coalesced b128 output stores
//  - uniform params via kernel-arg pointers -> SMEM scalar loads
//  - tail block (never taken for B=2^21): direct predicated global loads, no LDS/barrier

#ifndef USE_ASYNC_LDS
#define USE_ASYNC_LDS 1
#endif

#define BLK 256
#define RS_QUADS ((BLK * 13) / 4)   /* 832 float4s = 3*BLK + 64 */

__device__ __forceinline__ void async_b128_to_lds(const float* gaddr, const float* lds_ptr) {
    unsigned lds_off = (unsigned)(unsigned long long)(const void*)lds_ptr;
    asm volatile("global_load_async_to_lds_b128 %0, %1, off"
                 :: "v"(lds_off), "v"(gaddr) : "memory");
}

__global__ __launch_bounds__(BLK) void attctrl_kernel(
    const float* __restrict__ root_state,   // (n,13)
    const float* __restrict__ ctrl,         // (n,4)
    const float* __restrict__ mass_p,       // scalar
    const float* __restrict__ grav_p,       // scalar
    const float* __restrict__ mixer,        // (4,4) row-major
    const float* __restrict__ max_thrusts,  // (4,)
    const float* __restrict__ gA,           // (3,)
    const float* __restrict__ gR,           // (3,)
    float* __restrict__ out,                // (n,4)
    int n)
{
    __shared__ __align__(16) float s_rs[BLK * 13];
    __shared__ __align__(16) float s_ct[BLK * 4];

    const int tid  = threadIdx.x;
    const int base = blockIdx.x * BLK;
    const bool full = (n - base) >= BLK;
    const bool active = (base + tid) < n;

    float qw, qx, qy, qz, wx, wy, wz, roll, pitch, yawr, thr;

    if (full) {
        const float* g_rs = root_state + (long long)base * 13;
        const float* g_ct = ctrl + (long long)base * 4;
#if USE_ASYNC_LDS
        // 832 float4s of root_state: 3 full rounds + one 64-thread round; 1 float4 of ctrl.
        async_b128_to_lds(g_rs + (tid          ) * 4, &s_rs[(tid          ) * 4]);
        async_b128_to_lds(g_rs + (tid +     BLK) * 4, &s_rs[(tid +     BLK) * 4]);
        async_b128_to_lds(g_rs + (tid + 2 * BLK) * 4, &s_rs[(tid + 2 * BLK) * 4]);
        if (tid < RS_QUADS - 3 * BLK)
            async_b128_to_lds(g_rs + (tid + 3 * BLK) * 4, &s_rs[(tid + 3 * BLK) * 4]);
        async_b128_to_lds(g_ct + tid * 4, &s_ct[tid * 4]);
        asm volatile("s_wait_asynccnt 0" ::: "memory");
#else
        ((float4*)s_rs)[tid]           = ((const float4*)g_rs)[tid];
        ((float4*)s_rs)[tid + BLK]     = ((const float4*)g_rs)[tid + BLK];
        ((float4*)s_rs)[tid + 2 * BLK] = ((const float4*)g_rs)[tid + 2 * BLK];
        if (tid < RS_QUADS - 3 * BLK)
            ((float4*)s_rs)[tid + 3 * BLK] = ((const float4*)g_rs)[tid + 3 * BLK];
        ((float4*)s_ct)[tid] = ((const float4*)g_ct)[tid];
#endif
        __syncthreads();

        // per-element state (stride 13 -> conflict-free across 64 banks)
        const float* rs = &s_rs[tid * 13];
        qw = rs[3];  qx = rs[4];  qy = rs[5];  qz = rs[6];
        wx = rs[10]; wy = rs[11]; wz = rs[12];
        const float* ct = &s_ct[tid * 4];
        roll = ct[0]; pitch = ct[1]; yawr = ct[2]; thr = ct[3];
    } else {
        // tail block: direct predicated strided loads, no staging
        if (active) {
            const float* rs = root_state + (long long)(base + tid) * 13;
            const float* ct = ctrl + (long long)(base + tid) * 4;
            qw = rs[3];  qx = rs[4];  qy = rs[5];  qz = rs[6];
            wx = rs[10]; wy = rs[11]; wz = rs[12];
            roll = ct[0]; pitch = ct[1]; yawr = ct[2]; thr = ct[3];
        }
    }

    if (!active) return;

    // uniform params -> scalar (SMEM) loads
    const float mass = *mass_p, grav = *grav_p;
    const float gA0 = gA[0], gA1 = gA[1], gRz = gR[2];
    const float gR0 = gR[0], gR1 = gR[1];
    const float mt0 = max_thrusts[0], mt1 = max_thrusts[1];
    const float mt2 = max_thrusts[2], mt3 = max_thrusts[3];

    // R(q)
    const float xx = qx * qx, yy = qy * qy, zz = qz * qz;
    const float xy = qx * qy, xz = qx * qz, yz = qy * qz;
    const float qwx = qw * qx, qwy = qw * qy, qwz = qw * qz;
    const float r00 = 1.f - 2.f * (yy + zz), r01 = 2.f * (xy - qwz), r02 = 2.f * (xz + qwy);
    const float r10 = 2.f * (xy + qwz), r11 = 1.f - 2.f * (xx + zz), r12 = 2.f * (yz - qwx);
    const float r20 = 2.f * (xz - qwy), r21 = 2.f * (yz + qwx), r22 = 1.f - 2.f * (xx + yy);

    // R_des = Rz(yawr) * Rx(roll) * Ry(pitch), composed symbolically
    const float cy = __cosf(yawr), sy = __sinf(yawr);
    const float cr = __cosf(roll), sr = __sinf(roll);
    const float cp = __cosf(pitch), sp = __sinf(pitch);

    const float D00 = cy * cp - sy * sr * sp;
    const float D10 = sy * cp + cy * sr * sp;
    const float D20 = -cr * sp;
    const float D01 = -sy * cr;
    const float D11 = cy * cr;
    const float D21 = sr;
    const float D02 = cy * sp + sy * sr * cp;
    const float D12 = sy * sp - cy * sr * cp;
    const float D22 = cr * cp;

    // M = R_des^T * R ; only 5 entries needed
    const float M02 = D00 * r02 + D10 * r12 + D20 * r22;
    const float M12 = D01 * r02 + D11 * r12 + D21 * r22;
    const float M22 = D02 * r02 + D12 * r12 + D22 * r22;
    const float M21 = D02 * r01 + D12 * r11 + D22 * r21;
    const float M20 = D02 * r00 + D12 * r10 + D22 * r20;

    // E = 0.5(M - M^T): angle_error = (E21, E02, 0)
    const float aex = 0.5f * (M21 - M12);
    const float aey = 0.5f * (M02 - M20);

    // angular_rate_error = w - yawr * M[:,2]   (cross(w,w) == 0 exactly)
    const float rex = wx - yawr * M02;
    const float rey = wy - yawr * M12;
    const float rez = wz - yawr * M22;

    const float a0 = -aex * gA0 - rex * gR0;
    const float a1 = -aey * gA1 - rey * gR1;
    const float a2 = -rez * gRz;
    const float a3 = thr * mass * grav;

    // cmd = aat @ mixer.T ; out = cmd / max_thrusts * 2 - 1
    float4 o;
    o.x = (a0 * mixer[0]  + a1 * mixer[1]  + a2 * mixer[2]  + a3 * mixer[3])  / mt0 * 2.f - 1.f;
    o.y = (a0 * mixer[4]  + a1 * mixer[5]  + a2 * mixer[6]  + a3 * mixer[7])  / mt1 * 2.f - 1.f;
    o.z = (a0 * mixer[8]  + a1 * mixer[9]  + a2 * mixer[10] + a3 * mixer[11]) / mt2 * 2.f - 1.f;
    o.w = (a0 * mixer[12] + a1 * mixer[13] + a2 * mixer[14] + a3 * mixer[15]) / mt3 * 2.f - 1.f;

    ((float4*)out)[base + tid] = o;
}

extern "C" void kernel_launch(void* const* d_in, const int* in_sizes, int n_in,
                              void* d_out, int out_size, void* d_ws, size_t ws_size,
                              hipStream_t stream) {
    const float* root_state  = (const float*)d_in[0];
    const float* ctrl        = (const float*)d_in[1];
    const float* mass        = (const float*)d_in[2];
    const float* grav        = (const float*)d_in[3];
    const float* mixer       = (const float*)d_in[4];
    const float* max_thrusts = (const float*)d_in[5];
    const float* gain_att    = (const float*)d_in[6];
    const float* gain_rate   = (const float*)d_in[7];

    const int n = in_sizes[0] / 13;
    const int blocks = (n + BLK - 1) / BLK;

    attctrl_kernel<<<blocks, BLK, 0, stream>>>(
        root_state, ctrl, mass, grav, mixer, max_thrusts, gain_att, gain_rate,
        (float*)d_out, n);
}